// InitializerIndoc_30794915512481
// MI455X (gfx1250) — compile-verified
//
#include <hip/hip_runtime.h>
#include <stdint.h>

#define POOL_TR 8  // rows staged per async tile (double-buffered => 48KB LDS)

// ---------------------------------------------------------------------------
// gh[0:ncopy4) = hidden_states, gh[ncopy4:ntot4) = 0   (float4 granularity)
// Explicit branch (NOT a ternary) so the copy path lowers to
// global_load_b128 + global_store_b128 instead of a pointer-select through
// FLAT + a scratch-spilled zero vector (what round-1 codegen produced).
// (Placed first in the TU so the disasm snippet verifies this kernel.)
// ---------------------------------------------------------------------------
__global__ void ii_init_gh(float4* __restrict__ out, const float4* __restrict__ in,
                           int ncopy4, int ntot4) {
  int i = blockIdx.x * blockDim.x + threadIdx.x;
  if (i >= ntot4) return;
  if (i < ncopy4) {
    out[i] = in[i];
  } else {
    float4 z; z.x = z.y = z.z = z.w = 0.f;
    out[i] = z;
  }
}

// ---------------------------------------------------------------------------
// One block per target node: gather cnt source rows, mean, add onto gh[node].
// Source rows are staged global->LDS with CDNA5 async loads
// (global_load_async_to_lds_b128), double-buffered with s_wait_asynccnt.
// Each lane issues + consumes only its own 16B column chunk, so the per-wave
// ASYNCcnt is the only synchronization required (no barriers).
// blockDim.x == H/4 == 192 (6 wave32s).
// ---------------------------------------------------------------------------
__global__ __launch_bounds__(192) void ii_pool(float* __restrict__ gh,
    const int* __restrict__ offsets, const int* __restrict__ counts,
    const int* __restrict__ edge_list) {
  __shared__ float4 buf[2][POOL_TR][192];
  const int node = blockIdx.x;
  const int cnt = counts[node];
  if (cnt == 0) return;                       // uniform per block
  const int off = offsets[node];
  const int t = (int)threadIdx.x;
  const int h4 = (int)blockDim.x;             // 192

  const float4* ghv = (const float4*)gh;
  // generic LDS pointers carry the workgroup-relative LDS byte offset in the
  // low 32 bits; truncation yields the VDST value.
  uint32_t lbase = (uint32_t)(uintptr_t)(&buf[0][0][0]) + (uint32_t)(t * 16);

  auto issue_tile = [&](int bsel, int base) {
#pragma unroll
    for (int r = 0; r < POOL_TR; ++r) {
      int idx = base + r;
      idx = (idx < cnt) ? idx : (cnt - 1);    // clamp: keep ASYNCcnt bookkeeping fixed
      int srow = edge_list[off + idx];        // uniform -> SMEM (verified round 2)
      unsigned long long ga =
          (unsigned long long)(uintptr_t)(ghv + (size_t)srow * h4 + t);
      uint32_t la = lbase + (uint32_t)(((bsel * POOL_TR) + r) * 192 * 16);
      asm volatile("global_load_async_to_lds_b128 %0, %1, off"
                   :: "v"(la), "v"(ga) : "memory");
    }
  };

  const int ntiles = (cnt + POOL_TR - 1) / POOL_TR;
  issue_tile(0, 0);
  float4 acc; acc.x = acc.y = acc.z = acc.w = 0.f;
  for (int tile = 0; tile < ntiles; ++tile) {
    if (tile + 1 < ntiles) {
      issue_tile((tile + 1) & 1, (tile + 1) * POOL_TR);
      asm volatile("s_wait_asynccnt 8" ::: "memory");   // oldest tile complete
    } else {
      asm volatile("s_wait_asynccnt 0" ::: "memory");
    }
    int rows = cnt - tile * POOL_TR;
    if (rows > POOL_TR) rows = POOL_TR;
    for (int r = 0; r < rows; ++r) {
      float4 v = buf[tile & 1][r][t];
      acc.x += v.x; acc.y += v.y; acc.z += v.z; acc.w += v.w;
    }
  }
  float inv = 1.0f / (float)cnt;
  float4* out = (float4*)gh;
  size_t o = (size_t)node * h4 + t;
  float4 g = out[o];
  g.x += acc.x * inv; g.y += acc.y * inv; g.z += acc.z * inv; g.w += acc.w * inv;
  out[o] = g;
}

__global__ void ii_zero_i32(int* __restrict__ p, int n) {
  int i = blockIdx.x * blockDim.x + threadIdx.x;
  if (i < n) p[i] = 0;
}

__global__ void ii_count(const int* __restrict__ tgt, const int* __restrict__ typ,
                         int etype, int E, int* __restrict__ counts) {
  int e = blockIdx.x * blockDim.x + threadIdx.x;
  if (e < E && typ[e] == etype) atomicAdd(&counts[tgt[e]], 1);
}

// single-workgroup chunked Hillis-Steele exclusive scan (n ~ 70K, negligible)
__global__ __launch_bounds__(1024) void ii_scan(const int* __restrict__ counts,
                                                int* __restrict__ offsets, int n) {
  __shared__ int sh[1024];
  __shared__ int carry;
  if (threadIdx.x == 0) carry = 0;
  __syncthreads();
  for (int base = 0; base < n; base += 1024) {
    int i = base + (int)threadIdx.x;
    int v = (i < n) ? counts[i] : 0;
    sh[threadIdx.x] = v;
    __syncthreads();
    for (int d = 1; d < 1024; d <<= 1) {
      int add = (threadIdx.x >= (unsigned)d) ? sh[threadIdx.x - d] : 0;
      __syncthreads();
      sh[threadIdx.x] += add;
      __syncthreads();
    }
    if (i < n) offsets[i] = carry + sh[threadIdx.x] - v;  // exclusive
    __syncthreads();
    if (threadIdx.x == 1023) carry += sh[1023];
    __syncthreads();
  }
}

__global__ void ii_fill(const int* __restrict__ src, const int* __restrict__ tgt,
                        const int* __restrict__ typ, int etype, int E,
                        const int* __restrict__ offsets, int* __restrict__ cursor,
                        int* __restrict__ edge_list) {
  int e = blockIdx.x * blockDim.x + threadIdx.x;
  if (e < E && typ[e] == etype) {
    int tg = tgt[e];
    int p = atomicAdd(&cursor[tg], 1);
    edge_list[offsets[tg] + p] = src[e];
  }
}

// sort each segment's source list so float accumulation order is deterministic
// (atomic fill order is not). Segments are tiny (16 / 128 entries).
__global__ void ii_sort(const int* __restrict__ offsets, const int* __restrict__ counts,
                        int* __restrict__ edge_list, int n_total) {
  int node = blockIdx.x * blockDim.x + threadIdx.x;
  if (node >= n_total) return;
  int c = counts[node];
  if (c < 2) return;
  int* a = edge_list + offsets[node];
  for (int i = 1; i < c; ++i) {
    int key = a[i]; int j = i - 1;
    while (j >= 0 && a[j] > key) { a[j + 1] = a[j]; --j; }
    a[j + 1] = key;
  }
}

// paragraph rows <- first token row of each indoc chunk
__global__ void ii_para_init(float4* __restrict__ gh, int para_start,
                             int indoc, int h4) {
  int b = blockIdx.x, t = threadIdx.x;
  gh[(size_t)(para_start + b) * h4 + t] = gh[(size_t)b * indoc * h4 + t];
}

extern "C" void kernel_launch(void* const* d_in, const int* in_sizes, int n_in,
                              void* d_out, int out_size, void* d_ws, size_t ws_size,
                              hipStream_t stream) {
  (void)n_in; (void)ws_size;
  const float* hs  = (const float*)d_in[0];
  const int* esrc  = (const int*)d_in[2];
  const int* etgt  = (const int*)d_in[3];
  const int* etyp  = (const int*)d_in[4];
  float* gh = (float*)d_out;

  const int n_total = in_sizes[1];                  // 69761
  const int E       = in_sizes[2];                  // 65664
  const int H       = out_size / n_total;           // 768
  const int H4      = H / 4;                        // 192
  const long long S0 = (long long)in_sizes[0];
  const int max_token     = (int)(S0 / H);          // 65536
  const int max_paragraph = E - max_token;          // 128
  const int para_start    = n_total - 1 - max_paragraph;  // 69632
  const int max_indoc     = max_token / max_paragraph;    // 512

  int* counts    = (int*)d_ws;             // [n_total]
  int* cursor    = counts + n_total;       // [n_total] (contiguous with counts)
  int* offsets   = cursor + n_total;       // [n_total]
  int* edge_list = offsets + n_total;      // [E]

  const int ntot4  = n_total * H4;
  const int ncopy4 = max_token * H4;
  ii_init_gh<<<(ntot4 + 255) / 256, 256, 0, stream>>>(
      (float4*)gh, (const float4*)hs, ncopy4, ntot4);

  auto pool_pass = [&](int etype) {
    ii_zero_i32<<<(2 * n_total + 255) / 256, 256, 0, stream>>>(counts, 2 * n_total);
    ii_count<<<(E + 255) / 256, 256, 0, stream>>>(etgt, etyp, etype, E, counts);
    ii_scan<<<1, 1024, 0, stream>>>(counts, offsets, n_total);
    ii_fill<<<(E + 255) / 256, 256, 0, stream>>>(esrc, etgt, etyp, etype, E,
                                                 offsets, cursor, edge_list);
    ii_sort<<<(n_total + 255) / 256, 256, 0, stream>>>(offsets, counts,
                                                       edge_list, n_total);
    ii_pool<<<n_total, H4, 0, stream>>>(gh, offsets, counts, edge_list);
  };

  pool_pass(1);   // TOKEN_TO_SENTENCE mean pool
  ii_para_init<<<max_paragraph, H4, 0, stream>>>((float4*)gh, para_start,
                                                 max_indoc, H4);
  pool_pass(11);  // PARAGRAPH_TO_DOCUMENT mean pool
}